// NGCF_14302241096099
// MI455X (gfx1250) — compile-verified
//
#include <hip/hip_runtime.h>

typedef float v2f __attribute__((ext_vector_type(2)));
typedef float v8f __attribute__((ext_vector_type(8)));

#define DD 64          // feature dim
#define HS_PITCH 132   // 132 mod 64 == 4 -> conflict-free A-fragment reads
#define WAVES 6
#define ROWS (WAVES * 16)  // 96 nodes per block
#define TPB (WAVES * 32)   // 192 threads

// ---- degree counting: deg_out[src]++, deg_in[dst]++ ----
__global__ void ngcf_degree(const int* __restrict__ src, const int* __restrict__ dst,
                            int* __restrict__ dout, int* __restrict__ din, int E) {
  int g = blockIdx.x * blockDim.x + threadIdx.x;
  if (g < E) {
    atomicAdd(&dout[src[g]], 1);
    atomicAdd(&din[dst[g]], 1);
  }
}

// ---- per-node rsqrt of clipped degrees ----
__global__ void ngcf_rsqrt(const int* __restrict__ dout, const int* __restrict__ din,
                           float* __restrict__ sout, float* __restrict__ sin_, int N) {
  int g = blockIdx.x * blockDim.x + threadIdx.x;
  if (g < N) {
    int a = dout[g]; if (a < 1) a = 1;
    int b = din[g];  if (b < 1) b = 1;
    sout[g] = rsqrtf((float)a);
    sin_[g] = rsqrtf((float)b);
  }
}

// ---- edge scatter: p[dst] += rsqrt(deg_out[src]) * x[src]; 16 threads/edge, float4 each ----
__global__ void ngcf_scatter(const float* __restrict__ x, const int* __restrict__ src,
                             const int* __restrict__ dst, const float* __restrict__ sout,
                             float* __restrict__ p, int E) {
  int g = blockIdx.x * blockDim.x + threadIdx.x;
  int e = g >> 4;
  int c = g & 15;
  if (e < E) {
    int s = src[e];
    int d = dst[e];
    float w = sout[s];
    float4 v = ((const float4*)(x + (size_t)s * DD))[c];
    float* pd = p + (size_t)d * DD + (c << 2);
    atomicAdd(pd + 0, w * v.x);
    atomicAdd(pd + 1, w * v.y);
    atomicAdd(pd + 2, w * v.z);
    atomicAdd(pd + 3, w * v.w);
  }
}

// ---- per-node GEMM: out = (x + p') @ W1 + (x .* p') @ W2, p' = sin[node] * p[node]
//      via fp32 WMMA 16x16x4 over concatenated K=128 ----
__global__ __launch_bounds__(TPB) void ngcf_gemm(
    const float* __restrict__ x, const float* __restrict__ p,
    const float* __restrict__ sin_, const float* __restrict__ W1,
    const float* __restrict__ W2, float* __restrict__ out, int N, int do_relu) {
  __shared__ __align__(16) float hs[ROWS * HS_PITCH];  // H = [x+p' | x.*p'] per row

  const int tid = threadIdx.x;
  const int base = blockIdx.x * ROWS;

  // Stage H rows into LDS (96 rows x 16 float4 chunks = 1536 chunks / 192 threads)
  for (int i = tid; i < ROWS * 16; i += TPB) {
    int row = i >> 4;
    int c = (i & 15) << 2;
    int node = base + row;
    if (node >= N) node = N - 1;  // clamp; duplicate rows never stored
    float4 xv = *(const float4*)(x + (size_t)node * DD + c);
    float4 pv = *(const float4*)(p + (size_t)node * DD + c);
    float r = sin_[node];
    float4 pp = make_float4(pv.x * r, pv.y * r, pv.z * r, pv.w * r);
    float* h1 = &hs[row * HS_PITCH + c];
    float* h2 = &hs[row * HS_PITCH + DD + c];
    *(float4*)h1 = make_float4(xv.x + pp.x, xv.y + pp.y, xv.z + pp.z, xv.w + pp.w);
    *(float4*)h2 = make_float4(xv.x * pp.x, xv.y * pp.y, xv.z * pp.z, xv.w * pp.w);
  }
  __syncthreads();

  const int wave = tid >> 5;
  const int lane = tid & 31;
  const int m = lane & 15;
  const int khalf = (lane >> 4) << 1;  // lanes 0-15: K+0/K+1; lanes 16-31: K+2/K+3
  const float* hrow = &hs[(wave * 16 + m) * HS_PITCH];

  v8f acc0 = {}, acc1 = {}, acc2 = {}, acc3 = {};
  for (int kb = 0; kb < 2 * DD; kb += 4) {
    // A fragment (16x4 fp32): conflict-free b64 LDS read
    v2f a = *(const v2f*)(hrow + kb + khalf);
    // B fragment (4x16 fp32) from global (W resident in WGP$/L2); rows K and K+1
    const float* wr = (kb < DD) ? (W1 + (size_t)(kb + khalf) * DD)
                                : (W2 + (size_t)(kb + khalf - DD) * DD);
    v2f b0, b1, b2, b3;
    b0.x = wr[m];      b0.y = wr[DD + m];
    b1.x = wr[16 + m]; b1.y = wr[DD + 16 + m];
    b2.x = wr[32 + m]; b2.y = wr[DD + 32 + m];
    b3.x = wr[48 + m]; b3.y = wr[DD + 48 + m];
    acc0 = __builtin_amdgcn_wmma_f32_16x16x4_f32(false, a, false, b0, (short)0, acc0, false, false);
    acc1 = __builtin_amdgcn_wmma_f32_16x16x4_f32(false, a, false, b1, (short)0, acc1, false, false);
    acc2 = __builtin_amdgcn_wmma_f32_16x16x4_f32(false, a, false, b2, (short)0, acc2, false, false);
    acc3 = __builtin_amdgcn_wmma_f32_16x16x4_f32(false, a, false, b3, (short)0, acc3, false, false);
  }

  // C/D layout: VGPR r -> row r (lanes 0-15) / row r+8 (lanes 16-31), col = lane&15
  const int rbase = base + wave * 16 + ((lane >> 4) << 3);
  for (int r = 0; r < 8; ++r) {
    int node = rbase + r;
    if (node < N) {
      float v0 = acc0[r], v1 = acc1[r], v2 = acc2[r], v3 = acc3[r];
      if (do_relu) {
        v0 = fmaxf(v0, 0.f); v1 = fmaxf(v1, 0.f);
        v2 = fmaxf(v2, 0.f); v3 = fmaxf(v3, 0.f);
      }
      float* orow = out + (size_t)node * DD;
      orow[m] = v0; orow[16 + m] = v1; orow[32 + m] = v2; orow[48 + m] = v3;
    }
  }
}

extern "C" void kernel_launch(void* const* d_in, const int* in_sizes, int n_in,
                              void* d_out, int out_size, void* d_ws, size_t ws_size,
                              hipStream_t stream) {
  const float* x   = (const float*)d_in[0];
  const int*   src = (const int*)d_in[1];
  const int*   dst = (const int*)d_in[2];
  const float* W1a = (const float*)d_in[3];
  const float* W2a = (const float*)d_in[4];
  const float* W1b = (const float*)d_in[5];
  const float* W2b = (const float*)d_in[6];
  float* out = (float*)d_out;

  const int N = in_sizes[0] / DD;
  const int E = in_sizes[1];

  // workspace layout
  int* deg_out = (int*)d_ws;
  int* deg_in  = deg_out + N;
  float* sout  = (float*)(deg_in + N);
  float* sin_  = sout + N;
  float* p     = sin_ + N;  // N*DD floats, 16B-aligned (16*N bytes offset)

  // degrees (shared by both layers)
  hipMemsetAsync(deg_out, 0, (size_t)2 * N * sizeof(int), stream);
  ngcf_degree<<<(E + 255) / 256, 256, 0, stream>>>(src, dst, deg_out, deg_in, E);
  ngcf_rsqrt<<<(N + 255) / 256, 256, 0, stream>>>(deg_out, deg_in, sout, sin_, N);

  const int gemm_blocks = (N + ROWS - 1) / ROWS;
  const long long sc_items = (long long)E * 16;
  const int sc_blocks = (int)((sc_items + 255) / 256);

  // ---- Layer 1: h = relu((x+p')@W1_1 + (x.*p')@W2_1) -> d_out ----
  hipMemsetAsync(p, 0, (size_t)N * DD * sizeof(float), stream);
  ngcf_scatter<<<sc_blocks, 256, 0, stream>>>(x, src, dst, sout, p, E);
  ngcf_gemm<<<gemm_blocks, TPB, 0, stream>>>(x, p, sin_, W1a, W2a, out, N, 1);

  // ---- Layer 2: out = (h+p')@W1_2 + (h.*p')@W2_2, in place on d_out ----
  hipMemsetAsync(p, 0, (size_t)N * DD * sizeof(float), stream);
  ngcf_scatter<<<sc_blocks, 256, 0, stream>>>(out, src, dst, sout, p, E);
  ngcf_gemm<<<gemm_blocks, TPB, 0, stream>>>(out, p, sin_, W1b, W2b, out, N, 0);
}